// PointToPlaneLoss_70729521430590
// MI455X (gfx1250) — compile-verified
//
#include <hip/hip_runtime.h>
#include <hip/hip_bf16.h>
#include <math.h>

// Point-to-plane loss, MI455X (gfx1250).
// Scatter/atomic-bound problem (~120 MB working set, fits in 192 MB L2;
// ~0.25 GFLOP -> no WMMA role). CDNA5 features used:
//   - GLOBAL_LOAD_ASYNC_TO_LDS_B32 + s_wait_asynccnt: stage face-index tiles
//     into LDS once, reuse across all 32 batches.
//   - global_atomic_add_f32 (unsafeAtomicAdd): 57.6M non-returning FP32
//     scatter-adds, no CAS loop.
//   - wave32 shuffle reductions.

#define NBATCH 32
#define TILE_F 256
#define EPSF   1e-6f

#if defined(__has_builtin)
#if __has_builtin(__builtin_amdgcn_global_load_async_to_lds_b32)
#define HAS_ASYNC_LDS 1
#endif
#endif

typedef __attribute__((address_space(1))) int gint_t;
typedef __attribute__((address_space(3))) int lint_t;

__device__ __forceinline__ void wait_async_zero() {
#if defined(__has_builtin)
#if __has_builtin(__builtin_amdgcn_s_wait_asynccnt)
    __builtin_amdgcn_s_wait_asynccnt(0);
#else
    asm volatile("s_wait_asynccnt 0" ::: "memory");
#endif
#else
    asm volatile("s_wait_asynccnt 0" ::: "memory");
#endif
}

__device__ __forceinline__ void atomic_add_f32(float* p, float v) {
    // Lowers to global_atomic_add_f32 (non-returning) on gfx1250.
    unsafeAtomicAdd(p, v);
}

// ---------------------------------------------------------------------------
// Kernel 1: per face tile, compute face normals for all 32 batches and
// scatter-add into the per-vertex normal accumulator.
// Face indices for the tile are staged to LDS asynchronously and reused 32x.
// ---------------------------------------------------------------------------
__global__ __launch_bounds__(TILE_F) void face_normal_scatter(
    const float* __restrict__ gt,      // [B, V, 3]
    const int*   __restrict__ faces,   // [F, 3] (int32)
    float*       __restrict__ normals, // [B, V, 3] accumulator (pre-zeroed)
    int V, int F)
{
    __shared__ int s_idx[TILE_F * 3];

    const int t        = threadIdx.x;
    const int faceBase = blockIdx.x * TILE_F;
    const int nFaces   = (F - faceBase) < TILE_F ? (F - faceBase) : TILE_F;
    const int nElems   = nFaces * 3;

    const int* gsrc = faces + (size_t)faceBase * 3;

#ifdef HAS_ASYNC_LDS
    for (int k = t; k < nElems; k += TILE_F) {
        __builtin_amdgcn_global_load_async_to_lds_b32(
            (gint_t*)(gsrc + k), (lint_t*)(&s_idx[k]), 0, 0);
    }
    wait_async_zero();
#else
    for (int k = t; k < nElems; k += TILE_F)
        s_idx[k] = gsrc[k];
#endif
    __syncthreads();

    if (t >= nFaces) return;

    const int i0 = s_idx[t * 3 + 0] * 3;
    const int i1 = s_idx[t * 3 + 1] * 3;
    const int i2 = s_idx[t * 3 + 2] * 3;

#pragma unroll 1
    for (int b = 0; b < NBATCH; ++b) {
        const size_t boff = (size_t)b * V * 3;
        const float* g = gt + boff;

        const float ax = g[i0 + 0], ay = g[i0 + 1], az = g[i0 + 2];
        const float bx = g[i1 + 0], by = g[i1 + 1], bz = g[i1 + 2];
        const float cx = g[i2 + 0], cy = g[i2 + 1], cz = g[i2 + 2];

        const float e1x = bx - ax, e1y = by - ay, e1z = bz - az;
        const float e2x = cx - ax, e2y = cy - ay, e2z = cz - az;

        float nx = e1y * e2z - e1z * e2y;
        float ny = e1z * e2x - e1x * e2z;
        float nz = e1x * e2y - e1y * e2x;

        const float len = sqrtf(nx * nx + ny * ny + nz * nz);
        const float inv = 1.0f / (len + EPSF);
        nx *= inv; ny *= inv; nz *= inv;

        float* nb = normals + boff;
        atomic_add_f32(&nb[i0 + 0], nx);
        atomic_add_f32(&nb[i0 + 1], ny);
        atomic_add_f32(&nb[i0 + 2], nz);
        atomic_add_f32(&nb[i1 + 0], nx);
        atomic_add_f32(&nb[i1 + 1], ny);
        atomic_add_f32(&nb[i1 + 2], nz);
        atomic_add_f32(&nb[i2 + 0], nx);
        atomic_add_f32(&nb[i2 + 1], ny);
        atomic_add_f32(&nb[i2 + 2], nz);
    }
}

// ---------------------------------------------------------------------------
// Kernel 2: per vertex: normalize accumulated normal, |dot(pred-gt, n)|,
// block-reduce (wave32 shuffles + LDS across 8 waves), atomicAdd pre-scaled
// partial into the scalar output.
// ---------------------------------------------------------------------------
__global__ __launch_bounds__(256) void point_to_plane_reduce(
    const float* __restrict__ pred,    // [B, V, 3]
    const float* __restrict__ gt,      // [B, V, 3]
    const float* __restrict__ normals, // [B, V, 3]
    float*       __restrict__ out,     // scalar (pre-zeroed)
    int total,                         // B * V
    float scale)                       // 1 / (B*V + 1e-6)
{
    const int i = blockIdx.x * 256 + threadIdx.x;

    float val = 0.0f;
    if (i < total) {
        const size_t o = (size_t)i * 3;
        const float nx = normals[o + 0];
        const float ny = normals[o + 1];
        const float nz = normals[o + 2];
        const float len = sqrtf(nx * nx + ny * ny + nz * nz);
        const float inv = 1.0f / (len + EPSF);
        const float dx = pred[o + 0] - gt[o + 0];
        const float dy = pred[o + 1] - gt[o + 1];
        const float dz = pred[o + 2] - gt[o + 2];
        val = fabsf((dx * nx + dy * ny + dz * nz) * inv);
    }

    // wave32 reduction
    for (int off = 16; off > 0; off >>= 1)
        val += __shfl_down(val, off, 32);

    __shared__ float s_red[8];
    const int lane = threadIdx.x & 31;
    const int wid  = threadIdx.x >> 5;
    if (lane == 0) s_red[wid] = val;
    __syncthreads();

    if (wid == 0) {
        float v = (lane < 8) ? s_red[lane] : 0.0f;
        for (int off = 4; off > 0; off >>= 1)
            v += __shfl_down(v, off, 32);
        if (lane == 0)
            atomic_add_f32(out, v * scale);
    }
}

// ---------------------------------------------------------------------------
extern "C" void kernel_launch(void* const* d_in, const int* in_sizes, int n_in,
                              void* d_out, int out_size, void* d_ws, size_t ws_size,
                              hipStream_t stream)
{
    const float* pred  = (const float*)d_in[0];
    const float* gt    = (const float*)d_in[1];
    const int*   faces = (const int*)d_in[2];   // int -> const int* per harness
    float*       out   = (float*)d_out;

    const int B = NBATCH;
    const int V = in_sizes[0] / (B * 3);
    const int F = in_sizes[2] / 3;

    float* normals = (float*)d_ws;
    const size_t normalsBytes = (size_t)B * V * 3 * sizeof(float);

    (void)hipMemsetAsync(normals, 0, normalsBytes, stream);
    (void)hipMemsetAsync(out, 0, sizeof(float), stream);

    const int gridF = (F + TILE_F - 1) / TILE_F;
    face_normal_scatter<<<gridF, TILE_F, 0, stream>>>(gt, faces, normals, V, F);

    const int total = B * V;
    const float scale = (float)(1.0 / ((double)total + 1e-6));
    const int gridV = (total + 255) / 256;
    point_to_plane_reduce<<<gridV, 256, 0, stream>>>(pred, gt, normals, out,
                                                     total, scale);
}